// DeformableConv1d_60833916781079
// MI455X (gfx1250) — compile-verified
//
#include <hip/hip_runtime.h>

// Problem constants (from reference)
#define NB    16
#define CIN   64
#define LL    8192
#define COUT  64
#define LP1   (LL + 1)   // last valid padded index

typedef __attribute__((ext_vector_type(2))) float v2f;
typedef __attribute__((ext_vector_type(8))) float v8f;

// ---------------------------------------------------------------------------
// Kernel 1: offset conv.  off_ws[(n*3 + k)*LL + l] = conv(x, w_off)[n, 2k, l] + b_off[2k]
// One thread per (n, l). w_off channels 0,2,4 staged in LDS (3*192 floats).
// ---------------------------------------------------------------------------
__global__ void __launch_bounds__(256)
offconv_kernel(const float* __restrict__ x,
               const float* __restrict__ w_off,
               const float* __restrict__ b_off,
               float* __restrict__ off_ws) {
    __shared__ float sw[3 * 192];
    for (int t = threadIdx.x; t < 576; t += 256) {
        int k = t / 192, r = t - k * 192;
        sw[t] = w_off[(2 * k) * 192 + r];   // w_off[(2k)][ci][t] flat
    }
    __syncthreads();

    int id = blockIdx.x * 256 + threadIdx.x;   // 16*8192 total
    int n  = id >> 13;
    int l  = id & (LL - 1);

    const float* xn = x + (size_t)n * CIN * LL;
    float acc0 = b_off[0], acc1 = b_off[2], acc2 = b_off[4];
    for (int ci = 0; ci < CIN; ++ci) {
        const float* row = xn + ci * LL;
        float xm = (l >= 1)      ? row[l - 1] : 0.f;
        float x0 = row[l];
        float xp = (l + 1 < LL)  ? row[l + 1] : 0.f;
        const float* w0 = sw +       ci * 3;
        const float* w1 = sw + 192 + ci * 3;
        const float* w2 = sw + 384 + ci * 3;
        acc0 += xm * w0[0] + x0 * w0[1] + xp * w0[2];
        acc1 += xm * w1[0] + x0 * w1[1] + xp * w1[2];
        acc2 += xm * w2[0] + x0 * w2[1] + xp * w2[2];
    }
    off_ws[(n * 3 + 0) * LL + l] = acc0;
    off_ws[(n * 3 + 1) * LL + l] = acc1;
    off_ws[(n * 3 + 2) * LL + l] = acc2;
}

// ---------------------------------------------------------------------------
// Kernel 2: fused bilinear-gather + GEMM via V_WMMA_F32_16X16X4_F32.
//   out[n, o, l2] = sum_{m=0..191} w[o, m] * XD[n, m, l2] + bias[o]
//   XD[n, kk*64+j, p*64+c] = lerp of x_pad[n, c, left/right] at l = j*128 + p
// Block = 128 threads = 4 waves; each wave owns one 16-wide l2 tile and all
// 4 M-tiles (M=64).  W (64x192 f32 = 48KB) staged in LDS.
// Gathers are branch-free: addresses clamped into [1,L], results zeroed by
// cndmask when the padded index is out of range -> loads can batch per k-step.
// ---------------------------------------------------------------------------
__global__ void __launch_bounds__(128)
dconv_gemm_kernel(const float* __restrict__ x,
                  const float* __restrict__ w,
                  const float* __restrict__ bias,
                  const float* __restrict__ off_ws,
                  float* __restrict__ out) {
    __shared__ float sw[COUT * 192];   // 48 KB
    for (int t = threadIdx.x; t < COUT * 192; t += 128) sw[t] = w[t];
    __syncthreads();

    const int n    = blockIdx.x >> 7;        // 128 blocks per batch
    const int tg   = blockIdx.x & 127;
    const int wave = threadIdx.x >> 5;
    const int lane = threadIdx.x & 31;
    const int tile = tg * 4 + wave;          // 512 tiles of 16 columns
    const int l2_0 = tile * 16;
    const int p    = l2_0 >> 6;              // l % 128 component (uniform per tile)
    const int c0   = l2_0 & 63;              // channel base (16-aligned)

    const int mrow = lane & 15;              // row-in-tile (A/M) == col-in-tile (B/N)
    const int hi   = lane >> 4;              // lane half selects K pair {0,1} vs {2,3}
    const int c    = c0 + mrow;              // this lane's channel (B column)

    // x_pad[n, c, idx] for idx in [1, L] == xrow[idx]
    const float* xrow = x + ((size_t)n * CIN + c) * LL - 1;
    const float* offp = off_ws + n * 3 * LL;

    v8f acc0 = {}, acc1 = {}, acc2 = {}, acc3 = {};

    for (int ks = 0; ks < 48; ++ks) {
        const int m0 = ks * 4;

        // ---- B fragment: rows m0+2*hi, m0+2*hi+1 at column c ----
        // m0h is even, so both rows share kk and l1 = l0 + 128.
        const int m0h = m0 + hi * 2;
        const int kk  = m0h >> 6;
        const int j   = m0h & 63;
        const int l0v = j * 128 + p;
        const float* op = offp + kk * LL + l0v;
        float off0 = op[0];
        float off1 = op[128];

        float g0 = fminf(fmaxf((float)(l0v + 1)   + off0, 0.f), (float)LP1);
        float g1 = fminf(fmaxf((float)(l0v + 129) + off1, 0.f), (float)LP1);
        float fl0 = floorf(g0), fl1 = floorf(g1);
        int   left0  = (int)fl0,        left1  = (int)fl1;
        float alpha0 = g0 - fl0,        alpha1 = g1 - fl1;
        int   right0 = min(left0 + 1, LP1);
        int   right1 = min(left1 + 1, LP1);

        // clamped, branch-free gathers (addresses always within x[n,c,0..L-1])
        float xl0 = xrow[min(max(left0, 1), LL)];
        float xr0 = xrow[min(right0, LL)];
        float xl1 = xrow[min(max(left1, 1), LL)];
        float xr1 = xrow[min(right1, LL)];
        xl0 = ((unsigned)(left0 - 1) < (unsigned)LL) ? xl0 : 0.f;  // left in [1,LL]?
        xr0 = (right0 <= LL) ? xr0 : 0.f;                          // right >= 1 always
        xl1 = ((unsigned)(left1 - 1) < (unsigned)LL) ? xl1 : 0.f;
        xr1 = (right1 <= LL) ? xr1 : 0.f;

        v2f bfrag;
        bfrag.x = (1.f - alpha0) * xl0 + alpha0 * xr0;
        bfrag.y = (1.f - alpha1) * xl1 + alpha1 * xr1;

        // ---- A fragments from LDS: w[o0+mrow, m0 + 2*hi (+1)] as aligned float2 ----
        const int aidx = (mrow * 192 + m0h) >> 1;             // v2f units; even by construction
        const v2f* swa = (const v2f*)sw;
        v2f a0 = swa[aidx];                                    // o0 = 0
        v2f a1 = swa[aidx + (16 * 192 >> 1)];                  // o0 = 16
        v2f a2 = swa[aidx + (32 * 192 >> 1)];                  // o0 = 32
        v2f a3 = swa[aidx + (48 * 192 >> 1)];                  // o0 = 48

        acc0 = __builtin_amdgcn_wmma_f32_16x16x4_f32(false, a0, false, bfrag, (short)0, acc0, false, false);
        acc1 = __builtin_amdgcn_wmma_f32_16x16x4_f32(false, a1, false, bfrag, (short)0, acc1, false, false);
        acc2 = __builtin_amdgcn_wmma_f32_16x16x4_f32(false, a2, false, bfrag, (short)0, acc2, false, false);
        acc3 = __builtin_amdgcn_wmma_f32_16x16x4_f32(false, a3, false, bfrag, (short)0, acc3, false, false);
    }

    // ---- Epilogue: C/D layout -> VGPR v, lane: row = v + 8*hi, col = lane&15 ----
    float* outn = out + (size_t)n * COUT * LL;
    const int col = l2_0 + mrow;
#pragma unroll
    for (int v = 0; v < 8; ++v) {
        int ra = v + 8 * hi;
        int o0 =  0 + ra, o1 = 16 + ra, o2 = 32 + ra, o3 = 48 + ra;
        outn[(size_t)o0 * LL + col] = acc0[v] + bias[o0];
        outn[(size_t)o1 * LL + col] = acc1[v] + bias[o1];
        outn[(size_t)o2 * LL + col] = acc2[v] + bias[o2];
        outn[(size_t)o3 * LL + col] = acc3[v] + bias[o3];
    }
}

// ---------------------------------------------------------------------------
extern "C" void kernel_launch(void* const* d_in, const int* in_sizes, int n_in,
                              void* d_out, int out_size, void* d_ws, size_t ws_size,
                              hipStream_t stream) {
    const float* x     = (const float*)d_in[0];   // (16, 64, 8192)
    const float* w_off = (const float*)d_in[1];   // (6, 64, 3)
    const float* b_off = (const float*)d_in[2];   // (6,)
    const float* w     = (const float*)d_in[3];   // (64, 192)
    const float* b     = (const float*)d_in[4];   // (64,)
    float* out = (float*)d_out;                   // (16, 64, 8192)
    float* off_ws = (float*)d_ws;                 // 16*3*8192 floats = 1.5 MB

    // Kernel 1: 16*8192 threads
    offconv_kernel<<<(NB * LL) / 256, 256, 0, stream>>>(x, w_off, b_off, off_ws);
    // Kernel 2: 16 batches * 128 blocks, 4 waves/block, 1 tile/wave
    dconv_gemm_kernel<<<NB * 128, 128, 0, stream>>>(x, w, b, off_ws, out);
}